// GradientBoostingLoss_21431886807356
// MI455X (gfx1250) — compile-verified
//
#include <hip/hip_runtime.h>
#include <hip/hip_bf16.h>
#include <math.h>
#include <stdint.h>

#define NUM_CLASSES 1000
#define TILE_COLS   1024                  // padded row length in LDS (floats)
#define ROWS_PER_BLOCK 8                  // 8 waves of 32 lanes per 256-thread block
#define ROW_BYTES   (NUM_CLASSES * 4)     // 4000
#define TILE_BYTES  (TILE_COLS * 4)       // 4096
#define GRID_BLOCKS 1024                  // grid-stride: 4 rows per wave at B=32768

// Insert v into a descending sorted 4-tuple held in registers.
__device__ __forceinline__ void topk_insert(float v, float& t0, float& t1, float& t2, float& t3) {
    if (v > t3) {
        if (v > t1) {
            if (v > t0) { t3 = t2; t2 = t1; t1 = t0; t0 = v; }
            else        { t3 = t2; t2 = t1; t1 = v; }
        } else {
            if (v > t2) { t3 = t2; t2 = v; }
            else        { t3 = v; }
        }
    }
}

// Async-DMA one 4000B row into a 4KB LDS buffer: 8 chunks x (32 lanes x 16B).
// Tail lanes of the last chunk clamp their *global* address back in-row; their
// LDS slots land in cols >= 1000, which the compute phase never reads.
__device__ __forceinline__ void issue_row_dma(const float* rp, uint32_t lds_buf, int lane) {
#pragma unroll
    for (int j = 0; j < 8; ++j) {
        uint32_t bo = (uint32_t)(j * 512 + lane * 16);
        uint64_t ga = (uint64_t)(uintptr_t)rp + ((bo < (uint32_t)ROW_BYTES) ? bo : 0u);
        uint32_t la = lds_buf + bo;
        asm volatile("global_load_async_to_lds_b128 %0, %1, off"
                     :
                     : "v"(la), "v"(ga)
                     : "memory");
    }
}

__global__ __launch_bounds__(256) void topk_loss_kernel(
        const float* __restrict__ logits,
        const long long* __restrict__ labels,
        float* __restrict__ row_loss,
        int B) {
    // Double-buffered: 8 waves x 2 buffers x 4KB = 64KB LDS.
    __shared__ __align__(16) float tile[ROWS_PER_BLOCK * 2 * TILE_COLS];

    const int wave   = threadIdx.x >> 5;    // 0..7
    const int lane   = threadIdx.x & 31;    // 0..31 (wave32)
    const int stride = (int)gridDim.x * ROWS_PER_BLOCK;
    int row = (int)blockIdx.x * ROWS_PER_BLOCK + wave;

    // LDS byte offset of this wave's two 4KB slabs (flat shared pointers carry
    // the workgroup-relative LDS offset in their low 32 bits).
    const uint32_t lds0 = (uint32_t)(uintptr_t)(&tile[0]) + (uint32_t)(wave * 2) * TILE_BYTES;

    // Prologue: start DMA for this wave's first row.
    if (row < B) issue_row_dma(logits + (size_t)row * NUM_CLASSES, lds0, lane);

    int cur = 0;
    for (; row < B; row += stride) {
        const int nrow = row + stride;
        if (nrow < B) {
            // Software pipeline: start next row's DMA into the other buffer,
            // then wait only for the *current* row's 8 transfers (ASYNCcnt
            // completes in order, so <=8 outstanding => oldest 8 are done).
            issue_row_dma(logits + (size_t)nrow * NUM_CLASSES,
                          lds0 + (uint32_t)((cur ^ 1) * TILE_BYTES), lane);
            asm volatile("s_wait_asynccnt 0x8" ::: "memory");
        } else {
            asm volatile("s_wait_asynccnt 0x0" ::: "memory");
        }

        const int label = (int)labels[row];
        const int lf = label >> 2;          // float4 chunk containing the label
        const int le = label & 3;           // element within that chunk

        float t0 = -INFINITY, t1 = -INFINITY, t2 = -INFINITY, t3 = -INFINITY;
        float lab = -INFINITY;

        const float4* wt4 =
            (const float4*)(&tile[(wave * 2 + cur) * TILE_COLS]);

        // 1000 = 250 float4 chunks exactly: chunk f valid iff f < 250.
        // Chunk index f = lane + 32j -> 2-way-interleaved banks, b128 loads.
#pragma unroll
        for (int j = 0; j < 8; ++j) {
            const int f = lane + (j << 5);
            if (f < (NUM_CLASSES / 4)) {
                const float4 v = wt4[f];
                if (f == lf) {
                    if (le == 0) lab = v.x; else topk_insert(v.x, t0, t1, t2, t3);
                    if (le == 1) lab = v.y; else topk_insert(v.y, t0, t1, t2, t3);
                    if (le == 2) lab = v.z; else topk_insert(v.z, t0, t1, t2, t3);
                    if (le == 3) lab = v.w; else topk_insert(v.w, t0, t1, t2, t3);
                } else {
                    topk_insert(v.x, t0, t1, t2, t3);
                    topk_insert(v.y, t0, t1, t2, t3);
                    topk_insert(v.z, t0, t1, t2, t3);
                    topk_insert(v.w, t0, t1, t2, t3);
                }
            }
        }

        // Butterfly merge of per-lane top-4 (and label logit) across the wave32.
#pragma unroll
        for (int m = 1; m < 32; m <<= 1) {
            float o0 = __shfl_xor(t0, m, 32);
            float o1 = __shfl_xor(t1, m, 32);
            float o2 = __shfl_xor(t2, m, 32);
            float o3 = __shfl_xor(t3, m, 32);
            float ol = __shfl_xor(lab, m, 32);
            lab = fmaxf(lab, ol);
            topk_insert(o0, t0, t1, t2, t3);
            topk_insert(o1, t0, t1, t2, t3);
            topk_insert(o2, t0, t1, t2, t3);
            topk_insert(o3, t0, t1, t2, t3);
        }

        if (lane == 0) {
            // loss = logsumexp([lab, t0..t3]) - lab, numerically stable.
            const float mx = fmaxf(lab, t0);        // t0 = max of the top-4
            const float s = expf(lab - mx) + expf(t0 - mx) + expf(t1 - mx)
                          + expf(t2 - mx) + expf(t3 - mx);
            row_loss[row] = mx + logf(s) - lab;
        }

        cur ^= 1;
    }
}

// Deterministic fixed-order reduction: one block, each thread sums a fixed
// strided slice, then an LDS tree reduce. No float atomics -> bitwise stable.
__global__ __launch_bounds__(256) void reduce_mean_kernel(
        const float* __restrict__ row_loss, float* __restrict__ out, int B) {
    __shared__ float sdata[256];
    float s = 0.0f;
    for (int i = threadIdx.x; i < B; i += 256) s += row_loss[i];
    sdata[threadIdx.x] = s;
    __syncthreads();
#pragma unroll
    for (int ofs = 128; ofs > 0; ofs >>= 1) {
        if ((int)threadIdx.x < ofs) sdata[threadIdx.x] += sdata[threadIdx.x + ofs];
        __syncthreads();
    }
    if (threadIdx.x == 0) out[0] = sdata[0] / (float)B;
}

extern "C" void kernel_launch(void* const* d_in, const int* in_sizes, int n_in,
                              void* d_out, int out_size, void* d_ws, size_t ws_size,
                              hipStream_t stream) {
    const float*     logits = (const float*)d_in[0];        // [B, 1000] f32
    const long long* labels = (const long long*)d_in[1];    // [B] i64
    const int B = in_sizes[1];                              // 32768

    float* row_loss = (float*)d_ws;                         // B floats scratch
    float* out      = (float*)d_out;                        // 1 float

    int blocks = (B + ROWS_PER_BLOCK - 1) / ROWS_PER_BLOCK;
    if (blocks > GRID_BLOCKS) blocks = GRID_BLOCKS;
    topk_loss_kernel<<<blocks, 256, 0, stream>>>(logits, labels, row_loss, B);
    reduce_mean_kernel<<<1, 256, 0, stream>>>(row_loss, out, B);
}